// MemLayer_10471130267974
// MI455X (gfx1250) — compile-verified
//
#include <hip/hip_runtime.h>
#include <hip/hip_bf16.h>
#include <math.h>

#define N_NODES 20000
#define DEG     16
#define NH      4
#define NC      16
#define DDIM    128
#define INDIM   256
#define SLOPE   0.2f
#define KBLK    64

typedef float v2f __attribute__((ext_vector_type(2)));
typedef float v8f __attribute__((ext_vector_type(8)));

__device__ __forceinline__ v8f wmma4(v2f a, v2f b, v8f c) {
  // D = A(16x4 f32) * B(4x16 f32) + C(16x16 f32)
  return __builtin_amdgcn_wmma_f32_16x16x4_f32(
      /*neg_a=*/false, a, /*neg_b=*/false, b,
      /*c_mod=*/(short)0, c, /*reuse_a=*/false, /*reuse_b=*/false);
}

__device__ __forceinline__ float leaky(float x) { return x >= 0.f ? x : SLOPE * x; }

// ---------------------------------------------------------------------------
// Out[row, col] = leaky( A[row,:K] . W[col,:K] + bias[col] ), Dout = 128.
// Block = 256 threads (8 waves); block computes a 16x128 tile.
// K-blocked: A tile (16xKBLK) and W slab (128xKBLK) staged via LDS with
// coalesced float4 bursts; fragments then read conflict-free (stride 68).
// grid.x = Nrows/16.
// ---------------------------------------------------------------------------
__global__ void gemm_bias_leaky(const float* __restrict__ A,
                                const float* __restrict__ W,
                                const float* __restrict__ bias,
                                float* __restrict__ Out, int K) {
  __shared__ __align__(16) float sA[16 * 68];
  __shared__ __align__(16) float sB[128 * 68];
  const int tid = threadIdx.x;
  const int lane = tid & 31;
  const int wave = tid >> 5;
  const int lr = lane & 15;
  const int hi = lane >> 4;      // 0: K=0,1   1: K=2,3
  const int kh = hi * 2;
  const int row0 = blockIdx.x * 16;
  const int col0 = wave * 16;
  v8f acc = {};
  for (int k0 = 0; k0 < K; k0 += KBLK) {
    __syncthreads();  // previous block's fragment reads done before restage
    {  // A tile: 16 rows x 64 cols; 256 threads x 1 float4 (coalesced)
      const int r = tid >> 4;
      const int c = (tid & 15) * 4;
      *(float4*)(&sA[r * 68 + c]) =
          *(const float4*)(A + (size_t)(row0 + r) * K + k0 + c);
    }
#pragma unroll
    for (int i = 0; i < 8; ++i) {  // W slab: 128 rows x 64 cols (coalesced)
      const int idx = tid + i * 256;
      const int r = idx >> 4;
      const int c = (idx & 15) * 4;
      *(float4*)(&sB[r * 68 + c]) =
          *(const float4*)(W + (size_t)r * K + k0 + c);
    }
    __syncthreads();
#pragma unroll
    for (int k = 0; k < KBLK; k += 4) {
      v2f a = *(const v2f*)(&sA[lr * 68 + k + kh]);
      v2f b = *(const v2f*)(&sB[(col0 + lr) * 68 + k + kh]);
      acc = wmma4(a, b, acc);
    }
  }
  const float bv = bias[col0 + lr];
#pragma unroll
  for (int r = 0; r < 8; ++r) {
    const int m = r + hi * 8;
    Out[(size_t)(row0 + m) * DDIM + col0 + lr] = leaky(acc[r] + bv);
  }
}

// ---------------------------------------------------------------------------
// out[row] = sum_c M[row][c]^2 ; one wave per row, cols multiple of 128.
// ---------------------------------------------------------------------------
__global__ void rowsq(const float* __restrict__ M, float* __restrict__ out,
                      int rows, int cols) {
  const int row = blockIdx.x * (blockDim.x >> 5) + (threadIdx.x >> 5);
  const int lane = threadIdx.x & 31;
  if (row >= rows) return;
  const float* p = M + (size_t)row * cols;
  float s = 0.f;
  for (int c = lane * 4; c < cols; c += 128) {
    float4 v = *(const float4*)(p + c);
    s += v.x * v.x + v.y * v.y + v.z * v.z + v.w * v.w;
  }
#pragma unroll
  for (int off = 16; off > 0; off >>= 1) s += __shfl_xor(s, off, 32);
  if (lane == 0) out[row] = s;
}

// ---------------------------------------------------------------------------
// cross = cent(64x128) @ h^T(128xN); q = 1/(1 + cn2 + hn2 - 2 cross),
// stored node-major: qt[n*64 + hc]. grid.x = N/16, blockDim=128 (4 waves = 4
// hc-tiles). The 16-node h tile is staged in LDS (coalesced, stride 132).
// ---------------------------------------------------------------------------
__global__ void qker(const float* __restrict__ cent, const float* __restrict__ h,
                     const float* __restrict__ hn2, const float* __restrict__ cn2,
                     float* __restrict__ qt) {
  __shared__ __align__(16) float sHt[16 * 132];
  const int tid = threadIdx.x;
  const int lane = tid & 31;
  const int wave = tid >> 5;
  const int lr = lane & 15;
  const int hi = lane >> 4;
  const int kh = hi * 2;
  const int hc0 = wave * 16;
  const int n0 = blockIdx.x * 16;
#pragma unroll
  for (int i = 0; i < 4; ++i) {  // 16 rows x 128 cols, 128 threads x 4 float4
    const int idx = tid + i * 128;
    const int r = idx >> 5;
    const int c = (idx & 31) * 4;
    *(float4*)(&sHt[r * 132 + c]) =
        *(const float4*)(h + (size_t)(n0 + r) * DDIM + c);
  }
  __syncthreads();
  const float* arow = cent + (size_t)(hc0 + lr) * DDIM;
  v8f acc = {};
#pragma unroll
  for (int k = 0; k < DDIM; k += 4) {
    v2f a = *(const v2f*)(arow + k + kh);
    v2f b = *(const v2f*)(&sHt[lr * 132 + k + kh]);  // B[k][n] = h[n][k]
    acc = wmma4(a, b, acc);
  }
  const float hn = hn2[n0 + lr];
#pragma unroll
  for (int r = 0; r < 8; ++r) {
    const int m = hc0 + r + hi * 8;
    const float dist = cn2[m] + hn - 2.f * acc[r];
    qt[(size_t)(n0 + lr) * 64 + m] = 1.f / (1.f + dist);
  }
}

// ---------------------------------------------------------------------------
// Per node (one wave): gather qn, normalize over neighbors, mix heads,
// softmax over neighbors -> w[16c x 16e]; t = w @ h_gather (16x128);
// out = leaky(t @ W21^T + b21) -> new_feat[node]. blockDim=64 (2 waves),
// grid.x = N/2.
// ---------------------------------------------------------------------------
__global__ void cluster_fused(const float* __restrict__ qt,
                              const int* __restrict__ adj,
                              const float* __restrict__ h,
                              const float* __restrict__ conv,
                              const float* __restrict__ W21,
                              const float* __restrict__ b21,
                              float* __restrict__ nf) {
  __shared__ __align__(16) float sH[2][16 * 136];  // gathered rows (K-major reads)
  __shared__ __align__(16) float sT[2][16 * 132];  // t = w @ hg (M-major reads)
  __shared__ __align__(16) float sW[2][16 * 17];   // w[c][e]
  __shared__ int sA[2][16];
  const int lane = threadIdx.x & 31;
  const int wid = threadIdx.x >> 5;
  const int lr = lane & 15;
  const int hi = lane >> 4;
  const int kh = hi * 2;
  const int node = blockIdx.x * 2 + wid;

  if (lane < DEG) sA[wid][lane] = adj[(size_t)node * DEG + lane];
  __syncthreads();

  // Stage the 16 gathered neighbor rows (coalesced float4 per row).
#pragma unroll 4
  for (int e = 0; e < DEG; ++e) {
    const float* hr = h + (size_t)sA[wid][e] * DDIM;
    float4 v = *(const float4*)(hr + lane * 4);
    *(float4*)(&sH[wid][e * 136 + lane * 4]) = v;
  }

  // w[c][e]: lane handles c = lr (both halves compute redundantly).
  {
    float wce[16];
#pragma unroll
    for (int e = 0; e < 16; ++e) wce[e] = 0.f;
#pragma unroll
    for (int hh = 0; hh < NH; ++hh) {
      float qv[16];
      float den = 0.f;
#pragma unroll
      for (int e = 0; e < 16; ++e) {
        qv[e] = qt[(size_t)sA[wid][e] * 64 + hh * 16 + lr];
        den += qv[e];
      }
      const float s = conv[hh] / den;
#pragma unroll
      for (int e = 0; e < 16; ++e) wce[e] += qv[e] * s;
    }
    float mx = wce[0];
#pragma unroll
    for (int e = 1; e < 16; ++e) mx = fmaxf(mx, wce[e]);
    float sum = 0.f;
#pragma unroll
    for (int e = 0; e < 16; ++e) { wce[e] = __expf(wce[e] - mx); sum += wce[e]; }
    const float inv = 1.f / sum;
    if (hi == 0) {
#pragma unroll
      for (int e = 0; e < 16; ++e) sW[wid][lr * 17 + e] = wce[e] * inv;
    }
  }
  __syncthreads();

  // GEMM1: t(16x128) = w(16x16) @ hg(16x128), K = e = 16.
  for (int dt = 0; dt < 8; ++dt) {
    v8f acc = {};
#pragma unroll
    for (int k = 0; k < 16; k += 4) {
      v2f a;
      a.x = sW[wid][lr * 17 + k + kh];
      a.y = sW[wid][lr * 17 + k + kh + 1];
      v2f b;
      b.x = sH[wid][(k + kh) * 136 + dt * 16 + lr];
      b.y = sH[wid][(k + kh + 1) * 136 + dt * 16 + lr];
      acc = wmma4(a, b, acc);
    }
#pragma unroll
    for (int r = 0; r < 8; ++r)
      sT[wid][(r + hi * 8) * 132 + dt * 16 + lr] = acc[r];
  }
  __syncthreads();

  // GEMM2: out(16x128) = leaky(t @ W21^T + b21), K = 128. W21 stays in
  // global: identical fragments for every wave on the device -> L1-resident.
  float* orow = nf + (size_t)node * (NC * DDIM);
  for (int dt = 0; dt < 8; ++dt) {
    v8f acc = {};
    const float* wrow = W21 + (size_t)(dt * 16 + lr) * DDIM;
#pragma unroll 4
    for (int k = 0; k < DDIM; k += 4) {
      v2f a = *(const v2f*)(&sT[wid][lr * 132 + k + kh]);
      v2f b = *(const v2f*)(wrow + k + kh);
      acc = wmma4(a, b, acc);
    }
    const float bv = b21[dt * 16 + lr];
#pragma unroll
    for (int r = 0; r < 8; ++r) {
      const int c = r + hi * 8;
      orow[(size_t)c * DDIM + dt * 16 + lr] = leaky(acc[r] + bv);
    }
  }
}

// ---------------------------------------------------------------------------
extern "C" void kernel_launch(void* const* d_in, const int* in_sizes, int n_in,
                              void* d_out, int out_size, void* d_ws, size_t ws_size,
                              hipStream_t stream) {
  const float* x_node = (const float*)d_in[0];
  const int*   adj    = (const int*)d_in[1];
  // d_in[2] = epoch (unused)
  const float* W1   = (const float*)d_in[3];
  const float* b1   = (const float*)d_in[4];
  const float* W2   = (const float*)d_in[5];
  const float* b2   = (const float*)d_in[6];
  const float* W21  = (const float*)d_in[7];
  const float* b21  = (const float*)d_in[8];
  const float* cent = (const float*)d_in[9];
  const float* conv = (const float*)d_in[10];

  float* out = (float*)d_out;
  float* h  = out;                          // h_prime  [N,128]
  float* nf = out + (size_t)N_NODES * DDIM; // new_feat [N,16,128]

  float* xb  = (float*)d_ws;                // [N,128] layer-1 activations
  float* hn2 = xb + (size_t)N_NODES * DDIM; // [N]
  float* cn2 = hn2 + N_NODES;               // [64]
  float* qtb = cn2 + 64;                    // [N,64] q node-major

  // x = leaky(x_node @ W1^T + b1)
  gemm_bias_leaky<<<N_NODES / 16, 256, 0, stream>>>(x_node, W1, b1, xb, INDIM);
  // h = leaky(x @ W2^T + b2)
  gemm_bias_leaky<<<N_NODES / 16, 256, 0, stream>>>(xb, W2, b2, h, DDIM);
  // norms
  rowsq<<<(N_NODES + 7) / 8, 256, 0, stream>>>(h, hn2, N_NODES, DDIM);
  rowsq<<<8, 256, 0, stream>>>(cent, cn2, NH * NC, DDIM);
  // q = 1/(1+dist), node-major
  qker<<<N_NODES / 16, 128, 0, stream>>>(cent, h, hn2, cn2, qtb);
  // fused cluster aggregation + final projection
  cluster_fused<<<N_NODES / 2, 64, 0, stream>>>(qtb, adj, h, conv, W21, b21, nf);
}